// ClassicalGCN_65481071395456
// MI455X (gfx1250) — compile-verified
//
#include <hip/hip_runtime.h>
#include <hip/hip_bf16.h>

typedef __attribute__((ext_vector_type(2))) float v2f;
typedef __attribute__((ext_vector_type(8))) float v8f;

// ---------------------------------------------------------------------------
// Utility kernels
// ---------------------------------------------------------------------------
__global__ void gcn_fill_f32(float* __restrict__ p, float v, size_t n) {
    size_t i = (size_t)blockIdx.x * blockDim.x + threadIdx.x;
    size_t stride = (size_t)gridDim.x * blockDim.x;
    for (; i < n; i += stride) p[i] = v;
}

__global__ void gcn_degree(const int* __restrict__ dst, float* __restrict__ deg, int E) {
    int e = blockIdx.x * blockDim.x + threadIdx.x;
    if (e < E) atomicAdd(&deg[dst[e]], 1.0f);
}

__global__ void gcn_rsqrt_inplace(float* __restrict__ d, int n) {
    int i = blockIdx.x * blockDim.x + threadIdx.x;
    if (i < n) {
        float v = d[i];
        d[i] = (v > 0.0f) ? __frsqrt_rn(v) : 0.0f;
    }
}

// ---------------------------------------------------------------------------
// Dense GEMM via full-precision f32 WMMA (V_WMMA_F32_16X16X4_F32).
// One wave computes a 16-row strip across ALL NOUT columns; the A fragment is
// loaded once per k-step and reused across NOUT/16 B tiles.
// A: [N, K] row-major, W: [K, NOUT] row-major, C: [N, NOUT] row-major.
// N must be a multiple of 16 (40000 = 2500*16). K multiple of 4.
// ---------------------------------------------------------------------------
template <int NOUT>
__global__ __launch_bounds__(256) void gcn_gemm_wmma_f32(
    const float* __restrict__ A, const float* __restrict__ W,
    float* __restrict__ C, int N, int K) {
    constexpr int NT = NOUT / 16;
    const int wave = threadIdx.x >> 5;
    const int lane = threadIdx.x & 31;
    const int row0 = (blockIdx.x * 8 + wave) * 16;
    if (row0 >= N) return;            // wave-uniform: EXEC stays all-ones below

    const int m = lane & 15;          // M (for A) / N (for B) within tile
    const int g = lane >> 4;          // lane half selects K sub-pair
    const float* __restrict__ arow = A + (size_t)(row0 + m) * K + 2 * g;

    v8f acc[NT];
#pragma unroll
    for (int t = 0; t < NT; ++t) acc[t] = (v8f){0, 0, 0, 0, 0, 0, 0, 0};

    for (int k = 0; k < K; k += 4) {
        // A fragment (16x4 f32): VGPR0 = K=k+2g, VGPR1 = K=k+2g+1
        v2f a = *(const v2f*)(arow + k);
        const float* __restrict__ wrow = W + (size_t)(k + 2 * g) * NOUT + m;
#pragma unroll
        for (int t = 0; t < NT; ++t) {
            v2f b;                    // B fragment (4x16 f32), mirrored layout
            b.x = wrow[t * 16];
            b.y = wrow[NOUT + t * 16];
            acc[t] = __builtin_amdgcn_wmma_f32_16x16x4_f32(
                false, a, false, b, (short)0, acc[t], false, false);
        }
    }

    // C/D layout: VGPR r -> row (r + 8*g), col = lane&15 within tile
#pragma unroll
    for (int t = 0; t < NT; ++t) {
        float* __restrict__ crow = C + (size_t)(row0 + 8 * g) * NOUT + t * 16 + m;
#pragma unroll
        for (int r = 0; r < 8; ++r) {
            crow[(size_t)r * NOUT] = acc[t][r];
        }
    }
}

// ---------------------------------------------------------------------------
// Edge scatter: out[dst, f] += h[src, f] * dinv[src] * dinv[dst]
// blockDim.x = F (features, coalesced), blockDim.y = edges per block.
// Whole working set sits in the 192MB L2, so these atomics resolve in-cache.
// ---------------------------------------------------------------------------
__global__ void gcn_scatter(const float* __restrict__ h, const int* __restrict__ src,
                            const int* __restrict__ dst, const float* __restrict__ dinv,
                            float* __restrict__ out, int E, int F) {
    int e = blockIdx.x * blockDim.y + threadIdx.y;
    if (e >= E) return;
    int s = src[e];
    int d = dst[e];
    float nrm = dinv[s] * dinv[d];
    int f = threadIdx.x;
    atomicAdd(&out[(size_t)d * F + f], h[(size_t)s * F + f] * nrm);
}

// ---------------------------------------------------------------------------
// Epilogue: agg += self-loop term + bias, optional ReLU (in place).
// ---------------------------------------------------------------------------
__global__ void gcn_epilogue(float* __restrict__ agg, const float* __restrict__ h,
                             const float* __restrict__ dinv, const float* __restrict__ bias,
                             size_t total, int F, int do_relu) {
    size_t i = (size_t)blockIdx.x * blockDim.x + threadIdx.x;
    if (i >= total) return;
    size_t node = i / F;
    int f = (int)(i - node * F);
    float di = dinv[node];
    float v = agg[i] + h[i] * di * di + bias[f];
    agg[i] = do_relu ? fmaxf(v, 0.0f) : v;
}

// ---------------------------------------------------------------------------
extern "C" void kernel_launch(void* const* d_in, const int* in_sizes, int n_in,
                              void* d_out, int out_size, void* d_ws, size_t ws_size,
                              hipStream_t stream) {
    const float* x  = (const float*)d_in[0];
    const int*   ei = (const int*)d_in[1];
    const float* W1 = (const float*)d_in[2];
    const float* b1 = (const float*)d_in[3];
    const float* W2 = (const float*)d_in[4];
    const float* b2 = (const float*)d_in[5];
    float* out = (float*)d_out;

    const int DIN = 128, DH = 128, DOUT = 64;
    const int N = in_sizes[0] / DIN;     // 40000
    const int E = in_sizes[1] / 2;       // 640000
    const int* src = ei;
    const int* dst = ei + E;

    // Workspace layout (floats): dinv | h (also reused for g) | agg1
    float* ws = (float*)d_ws;
    size_t dinvPad = ((size_t)N + 255) & ~(size_t)255;
    float* dinv = ws;
    float* h    = ws + dinvPad;                  // N*DH floats (g reuses this)
    float* agg1 = h + (size_t)N * DH;            // N*DH floats

    const int TB = 256;
    size_t nh = (size_t)N * DH;
    size_t no = (size_t)N * DOUT;

    // 1) init: deg=1 (self-loop), agg buffers = 0
    gcn_fill_f32<<<(int)((N + TB - 1) / TB), TB, 0, stream>>>(dinv, 1.0f, (size_t)N);
    gcn_fill_f32<<<2048, TB, 0, stream>>>(agg1, 0.0f, nh);
    gcn_fill_f32<<<2048, TB, 0, stream>>>(out, 0.0f, no);

    // 2) in-degree, 3) dinv = rsqrt(deg)
    gcn_degree<<<(E + TB - 1) / TB, TB, 0, stream>>>(dst, dinv, E);
    gcn_rsqrt_inplace<<<(N + TB - 1) / TB, TB, 0, stream>>>(dinv, N);

    // 4) h = x @ W1  (f32 WMMA)
    int gemmBlocks = ((N / 16) + 7) / 8;
    gcn_gemm_wmma_f32<DH><<<gemmBlocks, 256, 0, stream>>>(x, W1, h, N, DIN);

    // 5) layer-1 edge scatter
    {
        dim3 blk(DH, 2);
        gcn_scatter<<<(E + 1) / 2, blk, 0, stream>>>(h, src, dst, dinv, agg1, E, DH);
    }

    // 6) epilogue 1: + self-loop + b1, ReLU (agg1 becomes layer-2 input)
    gcn_epilogue<<<(int)((nh + TB - 1) / TB), TB, 0, stream>>>(agg1, h, dinv, b1, nh, DH, 1);

    // 7) g = relu(h1) @ W2  (f32 WMMA), reuse h buffer for g
    gcn_gemm_wmma_f32<DOUT><<<gemmBlocks, 256, 0, stream>>>(agg1, W2, h, N, DH);

    // 8) layer-2 edge scatter into d_out
    {
        dim3 blk(DOUT, 4);
        gcn_scatter<<<(E + 3) / 4, blk, 0, stream>>>(h, src, dst, dinv, out, E, DOUT);
    }

    // 9) epilogue 2: + self-loop + b2 (no ReLU)
    gcn_epilogue<<<(int)((no + TB - 1) / TB), TB, 0, stream>>>(out, h, dinv, b2, no, DOUT, 0);
}